// HarmonicLayerL1_44178033606950
// MI455X (gfx1250) — compile-verified
//
#include <hip/hip_runtime.h>
#include <hip/hip_bf16.h>

#define INPUT_DIM  50000
#define OUTPUT_DIM 2048
#define EMBED_DIM  64
#define HIDDEN_DIM 128

typedef float v2f __attribute__((ext_vector_type(2)));
typedef float v8f __attribute__((ext_vector_type(8)));

// ---------------------------------------------------------------------------
// Kernel 1: row-wise argmax over x [B, 50000]. Bandwidth-bound (204.8 MB).
// One block per row, float4 streaming loads, LDS tree reduction with
// first-occurrence tie-break (val equal -> smaller index wins).
// ---------------------------------------------------------------------------
__global__ __launch_bounds__(256) void argmax_kernel(const float* __restrict__ x,
                                                     int* __restrict__ out_idx) {
  const int row = blockIdx.x;
  const float4* __restrict__ xv =
      (const float4*)(x + (long)row * INPUT_DIM);

  float best = -__builtin_inff();
  int bidx = 0;
  // 50000 = 12500 float4 chunks
  for (int c = threadIdx.x; c < INPUT_DIM / 4; c += 256) {
    float4 v = xv[c];
    int base = c * 4;
    if (v.x > best) { best = v.x; bidx = base; }
    if (v.y > best) { best = v.y; bidx = base + 1; }
    if (v.z > best) { best = v.z; bidx = base + 2; }
    if (v.w > best) { best = v.w; bidx = base + 3; }
  }

  __shared__ float sval[256];
  __shared__ int   sidx[256];
  sval[threadIdx.x] = best;
  sidx[threadIdx.x] = bidx;
  __syncthreads();
  for (int s = 128; s > 0; s >>= 1) {
    if (threadIdx.x < (unsigned)s) {
      float ov = sval[threadIdx.x + s];
      int   oi = sidx[threadIdx.x + s];
      float mv = sval[threadIdx.x];
      int   mi = sidx[threadIdx.x];
      if (ov > mv || (ov == mv && oi < mi)) {
        sval[threadIdx.x] = ov;
        sidx[threadIdx.x] = oi;
      }
    }
    __syncthreads();
  }
  if (threadIdx.x == 0) out_idx[row] = sidx[0];
}

// ---------------------------------------------------------------------------
// Kernel 2: fused gather + MLP via V_WMMA_F32_16X16X4_F32.
//   e = emb[idx]                       [16 x 64]   (LDS)
//   hidden = relu(e @ W1 + b1)        [16 x 128]  (LDS)
//   feats  = hidden @ W2 + b2         [16 x 128]  (global ws)
// One workgroup = 4 waves handles a 16-row batch strip; each wave owns two
// 16x16 N-tiles. Bias folded into accumulator init, ReLU on registers.
// ---------------------------------------------------------------------------
__global__ __launch_bounds__(128) void mlp_kernel(const float* __restrict__ emb,
                                                  const float* __restrict__ W1,
                                                  const float* __restrict__ b1,
                                                  const float* __restrict__ W2,
                                                  const float* __restrict__ b2,
                                                  const int*   __restrict__ idx,
                                                  float* __restrict__ feats) {
  __shared__ float e_lds[16][EMBED_DIM];    // 4 KB
  __shared__ float h_lds[16][HIDDEN_DIM];   // 8 KB
  __shared__ int   idx_lds[16];

  const int tid  = threadIdx.x;   // 0..127
  const int lane = tid & 31;
  const int wave = tid >> 5;      // 0..3
  const int row0 = blockIdx.x * 16;

  if (tid < 16) idx_lds[tid] = idx[row0 + tid];
  __syncthreads();

  // Gather 16 embedding rows (16*64 floats) cooperatively.
  for (int i = tid; i < 16 * EMBED_DIM; i += 128) {
    int r = i >> 6;
    int c = i & 63;
    e_lds[r][c] = emb[(long)idx_lds[r] * EMBED_DIM + c];
  }
  __syncthreads();

  const int half = (lane < 16) ? 0 : 1;  // K-halves for f32 A/B frags
  const int ln   = lane & 15;

  // ---- GEMM1: hidden = relu(e @ W1 + b1), K = 64 -> 16 k-steps ----
  for (int t = 0; t < 2; ++t) {
    const int n0 = (2 * wave + t) * 16;
    v8f acc;
    const float bias = b1[n0 + ln];
    for (int r = 0; r < 8; ++r) acc[r] = bias;

    for (int ks = 0; ks < EMBED_DIM / 4; ++ks) {
      const int k0 = ks * 4 + half * 2;
      v2f a, b;
      a.x = e_lds[ln][k0 + 0];
      a.y = e_lds[ln][k0 + 1];
      b.x = W1[(k0 + 0) * HIDDEN_DIM + n0 + ln];
      b.y = W1[(k0 + 1) * HIDDEN_DIM + n0 + ln];
      acc = __builtin_amdgcn_wmma_f32_16x16x4_f32(
          false, a, false, b, (short)0, acc, false, false);
    }
    // ReLU + stage into LDS. D VGPR r -> row (r + half*8), col (n0 + ln).
    for (int r = 0; r < 8; ++r) {
      float v = acc[r];
      h_lds[r + half * 8][n0 + ln] = v > 0.0f ? v : 0.0f;
    }
  }
  __syncthreads();

  // ---- GEMM2: feats = hidden @ W2 + b2, K = 128 -> 32 k-steps ----
  for (int t = 0; t < 2; ++t) {
    const int n0 = (2 * wave + t) * 16;
    v8f acc;
    const float bias = b2[n0 + ln];
    for (int r = 0; r < 8; ++r) acc[r] = bias;

    for (int ks = 0; ks < HIDDEN_DIM / 4; ++ks) {
      const int k0 = ks * 4 + half * 2;
      v2f a, b;
      a.x = h_lds[ln][k0 + 0];
      a.y = h_lds[ln][k0 + 1];
      b.x = W2[(k0 + 0) * HIDDEN_DIM + n0 + ln];
      b.y = W2[(k0 + 1) * HIDDEN_DIM + n0 + ln];
      acc = __builtin_amdgcn_wmma_f32_16x16x4_f32(
          false, a, false, b, (short)0, acc, false, false);
    }
    for (int r = 0; r < 8; ++r) {
      feats[(long)(row0 + r + half * 8) * HIDDEN_DIM + n0 + ln] = acc[r];
    }
  }
}

// ---------------------------------------------------------------------------
// Kernel 3: L1 cdist + inverse-distance normalization, fused per batch row.
// weights (1 MB) stays L2-resident across the 1024 row-blocks.
// ---------------------------------------------------------------------------
__global__ __launch_bounds__(256) void dist_kernel(const float* __restrict__ feats,
                                                   const float* __restrict__ weights,
                                                   float* __restrict__ out) {
  const int row = blockIdx.x;
  __shared__ float f[HIDDEN_DIM];
  __shared__ float red[256];

  if (threadIdx.x < HIDDEN_DIM)
    f[threadIdx.x] = feats[(long)row * HIDDEN_DIM + threadIdx.x];
  __syncthreads();

  float inv[OUTPUT_DIM / 256];  // 8 outputs per thread
  float psum = 0.0f;
  for (int j = 0; j < OUTPUT_DIM / 256; ++j) {
    const int o = threadIdx.x + 256 * j;
    const float4* __restrict__ wv =
        (const float4*)(weights + (long)o * HIDDEN_DIM);
    float d = 0.0f;
    for (int h = 0; h < HIDDEN_DIM / 4; ++h) {
      float4 w = wv[h];
      const int hb = h * 4;
      d += fabsf(f[hb + 0] - w.x) + fabsf(f[hb + 1] - w.y) +
           fabsf(f[hb + 2] - w.z) + fabsf(f[hb + 3] - w.w);
    }
    float iv = 1.0f / (d + 1e-8f);   // N_POW == 1
    inv[j] = iv;
    psum += iv;
  }

  red[threadIdx.x] = psum;
  __syncthreads();
  for (int s = 128; s > 0; s >>= 1) {
    if (threadIdx.x < (unsigned)s) red[threadIdx.x] += red[threadIdx.x + s];
    __syncthreads();
  }
  const float rtot = 1.0f / red[0];

  for (int j = 0; j < OUTPUT_DIM / 256; ++j) {
    out[(long)row * OUTPUT_DIM + threadIdx.x + 256 * j] = inv[j] * rtot;
  }
}

// ---------------------------------------------------------------------------
extern "C" void kernel_launch(void* const* d_in, const int* in_sizes, int n_in,
                              void* d_out, int out_size, void* d_ws, size_t ws_size,
                              hipStream_t stream) {
  const float* x       = (const float*)d_in[0];
  const float* emb     = (const float*)d_in[1];
  const float* W1      = (const float*)d_in[2];
  const float* b1      = (const float*)d_in[3];
  const float* W2      = (const float*)d_in[4];
  const float* b2      = (const float*)d_in[5];
  const float* weights = (const float*)d_in[6];
  float* out           = (float*)d_out;

  const int B = in_sizes[0] / INPUT_DIM;  // 1024

  // Workspace: [0, 8K) argmax indices, [8K, ...) feats [B x 128] f32.
  int*   ws_idx   = (int*)d_ws;
  float* ws_feats = (float*)((char*)d_ws + 8192);

  argmax_kernel<<<B, 256, 0, stream>>>(x, ws_idx);
  mlp_kernel<<<B / 16, 128, 0, stream>>>(emb, W1, b1, W2, b2, ws_idx, ws_feats);
  dist_kernel<<<B, 256, 0, stream>>>(ws_feats, weights, out);
}